// CrfRnn_26542897890013
// MI455X (gfx1250) — compile-verified
//
#include <hip/hip_runtime.h>
#include <hip/hip_bf16.h>

#define HH 96
#define WW 96
#define NPIX 9216            // 96*96
#define NTILE 576            // NPIX/16  (16x16 WMMA tiles per dim)
#define JCHUNKS 72           // 576 j-tiles / 8 per wave

typedef float v2f __attribute__((ext_vector_type(2)));
typedef float v8f __attribute__((ext_vector_type(8)));

#if defined(__has_builtin)
#if __has_builtin(__builtin_amdgcn_wmma_f32_16x16x4_f32)
#define HAS_WMMA_F32X4 1
#endif
#endif

static __device__ __forceinline__ void bf16x2(unsigned int u, float& lo, float& hi) {
  union { unsigned int i; float f; } a, b;
  a.i = u << 16;
  b.i = u & 0xffff0000u;
  lo = a.f; hi = b.f;
}

static __device__ __forceinline__ float wave_sum32(float v) {
#pragma unroll
  for (int m = 16; m >= 1; m >>= 1) v += __shfl_xor(v, m, 32);
  return v;
}

// ---------------------------------------------------------------------------
// Pass A: build Ksp/Kbi (bf16, row-major NxN) with f32 16x16x4 WMMA Gram tiles.
// One wave (32 lanes) per (i-tile, 8 j-tiles). K[i,j] = exp(G_ij - h_i - h_j).
// ---------------------------------------------------------------------------
__global__ __launch_bounds__(32) void build_kernels_kernel(
    const float* __restrict__ img,
    __hip_bfloat16* __restrict__ Ksp,
    __hip_bfloat16* __restrict__ Kbi)
{
  const float invG = 1.0f / 3.0f;    // 1/GAMMA
  const float invA = 1.0f / 160.0f;  // 1/ALPHA
  const float invB = 1.0f / 3.0f;    // 1/BETA
  const int lane = threadIdx.x;      // 0..31
  const int half = lane >> 4;        // K-half select for A/B fragments
  const int lm   = lane & 15;        // row (A) / col (B) within 16-tile
  const int it   = blockIdx.x;       // i tile
  const int jt0  = blockIdx.y * 8;   // first j tile of this wave

  // ---- A fragments: lane holds row M=lm, K elements {2*half, 2*half+1} ----
  const int ia = it * 16 + lm;
  const float ya = (float)(ia / WW), xa = (float)(ia - (ia / WW) * WW);
  const float rA = img[ia] * invB, gA = img[NPIX + ia] * invB, bA = img[2 * NPIX + ia] * invB;
  // spatial features: (y/g, x/g, 0, 0)
  v2f a_sp  = half ? (v2f){0.f, 0.f} : (v2f){ya * invG, xa * invG};
  // bilateral features: (y/a, x/a, r/b, g/b | b/b, 0, 0, 0)  (two K=4 chunks)
  v2f a_bi0 = half ? (v2f){rA, gA}   : (v2f){ya * invA, xa * invA};
  v2f a_bi1 = half ? (v2f){0.f, 0.f} : (v2f){bA, 0.f};

  // ---- per-output-row 0.5*|f|^2 ; C layout: VGPR r -> row it*16 + r + 8*half
  float hs[8], hb[8];
#pragma unroll
  for (int r = 0; r < 8; ++r) {
    const int ir = it * 16 + r + 8 * half;
    const float y = (float)(ir / WW), x = (float)(ir - (ir / WW) * WW);
    const float rr = img[ir] * invB, gg = img[NPIX + ir] * invB, bb = img[2 * NPIX + ir] * invB;
    const float ys = y * invG, xs = x * invG;
    hs[r] = 0.5f * (ys * ys + xs * xs);
    const float yb = y * invA, xb = x * invA;
    hb[r] = 0.5f * (yb * yb + xb * xb + rr * rr + gg * gg + bb * bb);
  }

  for (int jj = 0; jj < 8; ++jj) {
    const int jt = jt0 + jj;
    const int jb = jt * 16 + lm;
    const float yj = (float)(jb / WW), xj = (float)(jb - (jb / WW) * WW);
    const float rJ = img[jb] * invB, gJ = img[NPIX + jb] * invB, bJ = img[2 * NPIX + jb] * invB;
    v2f b_sp  = half ? (v2f){0.f, 0.f} : (v2f){yj * invG, xj * invG};
    v2f b_bi0 = half ? (v2f){rJ, gJ}   : (v2f){yj * invA, xj * invA};
    v2f b_bi1 = half ? (v2f){0.f, 0.f} : (v2f){bJ, 0.f};

    const float ysj = yj * invG, xsj = xj * invG;
    const float hsj = 0.5f * (ysj * ysj + xsj * xsj);
    const float ybj = yj * invA, xbj = xj * invA;
    const float hbj = 0.5f * (ybj * ybj + xbj * xbj + rJ * rJ + gJ * gJ + bJ * bJ);

    v8f csp = {0.f, 0.f, 0.f, 0.f, 0.f, 0.f, 0.f, 0.f};
    v8f cbi = {0.f, 0.f, 0.f, 0.f, 0.f, 0.f, 0.f, 0.f};
#if HAS_WMMA_F32X4
    csp = __builtin_amdgcn_wmma_f32_16x16x4_f32(false, a_sp,  false, b_sp,  (short)0, csp, false, false);
    cbi = __builtin_amdgcn_wmma_f32_16x16x4_f32(false, a_bi0, false, b_bi0, (short)0, cbi, false, false);
    cbi = __builtin_amdgcn_wmma_f32_16x16x4_f32(false, a_bi1, false, b_bi1, (short)0, cbi, false, false);
#else
    // VALU fallback (direct dot products, same C layout)
#pragma unroll
    for (int r = 0; r < 8; ++r) {
      const int ir = it * 16 + r + 8 * half;
      const float yi = (float)(ir / WW), xi = (float)(ir - (ir / WW) * WW);
      csp[r] = (yi * invG) * ysj + (xi * invG) * xsj;
      cbi[r] = (yi * invA) * ybj + (xi * invA) * xbj +
               (img[ir] * invB) * rJ + (img[NPIX + ir] * invB) * gJ +
               (img[2 * NPIX + ir] * invB) * bJ;
    }
#endif
#pragma unroll
    for (int r = 0; r < 8; ++r) {
      const int ir = it * 16 + r + 8 * half;
      const size_t off = (size_t)ir * NPIX + jb;
      Ksp[off] = __float2bfloat16(__expf(csp[r] - hs[r] - hsj));
      Kbi[off] = __float2bfloat16(__expf(cbi[r] - hb[r] - hbj));
    }
  }
}

// ---------------------------------------------------------------------------
// Pass B: deterministic row sums of the stored (bf16-rounded) kernels.
// One wave per row; lanes stream b128 chunks.
// ---------------------------------------------------------------------------
__global__ __launch_bounds__(256) void row_sums_kernel(
    const __hip_bfloat16* __restrict__ Ksp,
    const __hip_bfloat16* __restrict__ Kbi,
    float* __restrict__ Ssp, float* __restrict__ Sbi)
{
  const int lane = threadIdx.x & 31;
  const int j = blockIdx.x * 8 + (threadIdx.x >> 5);
  const uint4* rs = (const uint4*)(Ksp + (size_t)j * NPIX);
  const uint4* rb = (const uint4*)(Kbi + (size_t)j * NPIX);
  float ss = 0.f, sb = 0.f;
  for (int c = lane; c < NPIX / 8; c += 32) {
    const uint4 us = rs[c];
    const uint4 ub = rb[c];
    float a0, a1;
    bf16x2(us.x, a0, a1); ss += a0 + a1;
    bf16x2(us.y, a0, a1); ss += a0 + a1;
    bf16x2(us.z, a0, a1); ss += a0 + a1;
    bf16x2(us.w, a0, a1); ss += a0 + a1;
    bf16x2(ub.x, a0, a1); sb += a0 + a1;
    bf16x2(ub.y, a0, a1); sb += a0 + a1;
    bf16x2(ub.z, a0, a1); sb += a0 + a1;
    bf16x2(ub.w, a0, a1); sb += a0 + a1;
  }
  ss = wave_sum32(ss);
  sb = wave_sum32(sb);
  if (lane == 0) { Ssp[j] = ss; Sbi[j] = sb; }
}

// q0 = softmax(cur)[label 0] = sigmoid(cur0 - cur1)
__global__ void init_q_kernel(const float* __restrict__ logits, float* __restrict__ q0) {
  const int j = blockIdx.x * blockDim.x + threadIdx.x;
  if (j < NPIX) q0[j] = 1.0f / (1.0f + __expf(logits[NPIX + j] - logits[j]));
}

// ---------------------------------------------------------------------------
// Pass C (x5): y_sp[j] = sum_i Ksp[j,i]*q0[i] (symmetry), same for bilateral,
// then 2x2 mixing / compatibility update fused in lane 0.
// One wave per pixel j, streaming bf16 rows at b128 granularity.
// ---------------------------------------------------------------------------
__global__ __launch_bounds__(256) void crf_iter_kernel(
    const __hip_bfloat16* __restrict__ Ksp,
    const __hip_bfloat16* __restrict__ Kbi,
    const float* __restrict__ Ssp, const float* __restrict__ Sbi,
    const float* __restrict__ qin, const float* __restrict__ unary,
    const float* __restrict__ wsp, const float* __restrict__ wbi,
    const float* __restrict__ cmat,
    float* __restrict__ cur, float* __restrict__ qout)
{
  const int lane = threadIdx.x & 31;
  const int j = blockIdx.x * 8 + (threadIdx.x >> 5);
  const uint4* rs = (const uint4*)(Ksp + (size_t)j * NPIX);
  const uint4* rb = (const uint4*)(Kbi + (size_t)j * NPIX);
  const float4* qv = (const float4*)qin;
  float accs = 0.f, accb = 0.f;
  for (int c = lane; c < NPIX / 8; c += 32) {
    __builtin_prefetch(rs + c + 128, 0, 0);
    __builtin_prefetch(rb + c + 128, 0, 0);
    const uint4 us = rs[c];
    const uint4 ub = rb[c];
    const float4 q0 = qv[2 * c];
    const float4 q1 = qv[2 * c + 1];
    float e0, e1;
    bf16x2(us.x, e0, e1); accs = fmaf(e0, q0.x, accs); accs = fmaf(e1, q0.y, accs);
    bf16x2(us.y, e0, e1); accs = fmaf(e0, q0.z, accs); accs = fmaf(e1, q0.w, accs);
    bf16x2(us.z, e0, e1); accs = fmaf(e0, q1.x, accs); accs = fmaf(e1, q1.y, accs);
    bf16x2(us.w, e0, e1); accs = fmaf(e0, q1.z, accs); accs = fmaf(e1, q1.w, accs);
    bf16x2(ub.x, e0, e1); accb = fmaf(e0, q0.x, accb); accb = fmaf(e1, q0.y, accb);
    bf16x2(ub.y, e0, e1); accb = fmaf(e0, q0.z, accb); accb = fmaf(e1, q0.w, accb);
    bf16x2(ub.z, e0, e1); accb = fmaf(e0, q1.x, accb); accb = fmaf(e1, q1.y, accb);
    bf16x2(ub.w, e0, e1); accb = fmaf(e0, q1.z, accb); accb = fmaf(e1, q1.w, accb);
  }
  accs = wave_sum32(accs);
  accb = wave_sum32(accb);
  if (lane == 0) {
    const float Ss = Ssp[j], Sb = Sbi[j];
    const float ns = 1.0f / (Ss + 1e-20f), nb = 1.0f / (Sb + 1e-20f);
    // label 0 from the matvec; label 1 via q1 = 1 - q0  =>  colsum - y.
    const float sp0 = accs * ns,      sp1 = (Ss - accs) * ns;
    const float bi0 = accb * nb,      bi1 = (Sb - accb) * nb;
    const float m0 = wsp[0] * sp0 + wsp[1] * sp1 + wbi[0] * bi0 + wbi[1] * bi1;
    const float m1 = wsp[2] * sp0 + wsp[3] * sp1 + wbi[2] * bi0 + wbi[3] * bi1;
    const float c0 = unary[j]        - (cmat[0] * m0 + cmat[1] * m1);
    const float c1 = unary[NPIX + j] - (cmat[2] * m0 + cmat[3] * m1);
    cur[j] = c0;
    cur[NPIX + j] = c1;
    qout[j] = 1.0f / (1.0f + __expf(c1 - c0));
  }
}

extern "C" void kernel_launch(void* const* d_in, const int* in_sizes, int n_in,
                              void* d_out, int out_size, void* d_ws, size_t ws_size,
                              hipStream_t stream) {
  (void)in_sizes; (void)n_in; (void)out_size; (void)ws_size;
  const float* image  = (const float*)d_in[0];  // [1,3,96,96]
  const float* logits = (const float*)d_in[1];  // [1,2,96,96] (= unary)
  const float* wsp    = (const float*)d_in[2];  // [2,2]
  const float* wbi    = (const float*)d_in[3];  // [2,2]
  const float* cmat   = (const float*)d_in[4];  // [2,2]
  float* out = (float*)d_out;                   // [1,2,96,96] = cur

  char* ws = (char*)d_ws;
  const size_t KSZ = (size_t)NPIX * NPIX * sizeof(__hip_bfloat16);  // ~170 MB each
  __hip_bfloat16* Ksp = (__hip_bfloat16*)ws;
  __hip_bfloat16* Kbi = (__hip_bfloat16*)(ws + KSZ);
  float* Ssp = (float*)(ws + 2 * KSZ);
  float* Sbi = Ssp + NPIX;
  float* qA  = Sbi + NPIX;
  float* qB  = qA + NPIX;

  // Build dense Gaussian kernels via f32 WMMA Gram tiles.
  build_kernels_kernel<<<dim3(NTILE, JCHUNKS), 32, 0, stream>>>(image, Ksp, Kbi);
  // Deterministic normalization sums.
  row_sums_kernel<<<NPIX / 8, 256, 0, stream>>>(Ksp, Kbi, Ssp, Sbi);
  // Initial soft assignment from the unary logits.
  init_q_kernel<<<(NPIX + 255) / 256, 256, 0, stream>>>(logits, qA);

  float* qin = qA;
  float* qout = qB;
  for (int t = 0; t < 5; ++t) {
    crf_iter_kernel<<<NPIX / 8, 256, 0, stream>>>(Ksp, Kbi, Ssp, Sbi, qin, logits,
                                                  wsp, wbi, cmat, out, qout);
    float* tmp = qin; qin = qout; qout = tmp;
  }
}